// ROSA_1bit_LAYER_56049323213581
// MI455X (gfx1250) — compile-verified
//
#include <hip/hip_runtime.h>
#include <stdint.h>

// ROSA 1-bit layer for MI455X (gfx1250).
//
// 256 independent suffix-automaton sequences (B=2, C=128, T=2048).
// One workgroup (64 threads = 2 wave32) per sequence:
//   phase 1: async global->LDS stage of the input column (CDNA5 async path:
//            global_load_async_to_lds_b32 + s_wait_asynccnt), bits in LDS
//   phase 2: lane 0 builds the suffix automaton with ALL state resident in LDS
//            (16-bit fields; 5 x 4098 x 2B ~= 41 KB << 320 KB/WGP).
//            Prediction walk is FUSED into the endpos-update walk (the update
//            walk provably covers the whole suffix chain each step, so the
//            predict walk is a strict prefix of it). `len>0` == `v != root`.
//   phase 3: all lanes do the elementwise {0, emb0, emb1} select + store.

#define T_LEN  2048
#define C_DIM  128
#define B_DIM  2
#define MAXS   (2 * T_LEN + 2)   // 4098 states max (z <= 2n+1 = 4097)
#define NONE16 0xFFFFu

__global__ __launch_bounds__(64) void rosa_sa_kernel(
    const float* __restrict__ x,     // [B,T,C]
    const float* __restrict__ emb0,  // [C]
    const float* __restrict__ emb1,  // [C]
    float* __restrict__ out)         // [B,T,C]
{
    __shared__ unsigned short tr0[MAXS];   // transition on symbol 0
    __shared__ unsigned short tr1[MAXS];   // transition on symbol 1
    __shared__ unsigned short lnk[MAXS];   // suffix link (NONE16 == -1)
    __shared__ unsigned short len_[MAXS];  // state length (kept for clone math)
    __shared__ unsigned short ep[MAXS];    // last end-position (NONE16 == -1)
    __shared__ float          xstage[T_LEN];
    __shared__ unsigned char  bits[T_LEN];
    __shared__ signed char    pred[T_LEN];

    const int seq = blockIdx.x;          // 0..255
    const int b   = seq / C_DIM;
    const int c   = seq % C_DIM;
    const int tid = threadIdx.x;

    // ---------------- phase 1: async-stage input column into LDS --------------
    const float* xcol = x + ((size_t)b * T_LEN) * C_DIM + c;  // element t at xcol[t*C]
    for (int t = tid; t < T_LEN; t += 64) {
        unsigned           lds_addr = (unsigned)(uintptr_t)(&xstage[t]);
        unsigned long long gaddr    = (unsigned long long)(uintptr_t)(xcol + (size_t)t * C_DIM);
        asm volatile("global_load_async_to_lds_b32 %0, %1, off"
                     :: "v"(lds_addr), "v"(gaddr)
                     : "memory");
    }
    asm volatile("s_wait_asynccnt 0" ::: "memory");
    __syncthreads();

    for (int t = tid; t < T_LEN; t += 64)
        bits[t] = (xstage[t] > 0.0f) ? 1 : 0;
    __syncthreads();

    // ---------------- phase 2: sequential suffix automaton (lane 0) ----------
    if (tid == 0) {
        tr0[0] = NONE16; tr1[0] = NONE16; lnk[0] = NONE16; len_[0] = 0; ep[0] = NONE16;
        int g = 0, z = 1;
        #pragma unroll 1
        for (int i = 0; i < T_LEN; ++i) {
            const int t = bits[i];
            unsigned short* __restrict__ trt = t ? tr1 : tr0;

            const int r = z++;
            tr0[r]  = NONE16;
            tr1[r]  = NONE16;
            ep[r]   = NONE16;
            len_[r] = (unsigned short)(len_[g] + 1);

            int p = g;
            while (p != -1 && trt[p] == NONE16) {
                trt[p] = (unsigned short)r;
                unsigned short l = lnk[p];
                p = (l == NONE16) ? -1 : (int)l;
            }
            if (p == -1) {
                lnk[r] = 0;
            } else {
                const int q = (int)trt[p];
                if ((int)len_[p] + 1 == (int)len_[q]) {
                    lnk[r] = (unsigned short)q;
                } else {
                    const int u = z++;
                    tr0[u]  = tr0[q];
                    tr1[u]  = tr1[q];
                    len_[u] = (unsigned short)(len_[p] + 1);
                    lnk[u]  = lnk[q];
                    ep[u]   = ep[q];
                    while (p != -1 && trt[p] == (unsigned short)q) {
                        trt[p] = (unsigned short)u;
                        unsigned short l = lnk[p];
                        p = (l == NONE16) ? -1 : (int)l;
                    }
                    lnk[q] = (unsigned short)u;
                    lnk[r] = (unsigned short)u;
                }
            }
            g = r;

            // Fused prediction + endpos update.
            // Reference does:  (a) walk chain, break at first v with d[v]>0 &&
            // e[v]>=0, predict bits[e[v]+1];  (b) walk chain setting e[v]=i
            // while e[v]<i.  Since every e value is <= i-1 at step i, walk (b)
            // always covers the FULL chain, so (a) is a strict prefix of (b):
            // do one walk, record the first pre-update hit.  d[v]>0 <=> v!=0.
            {
                int v = g;
                int a = -1;
                bool found = false;
                while (v != -1) {
                    const unsigned short e = ep[v];          // pre-update value
                    if (!found && v != 0 && e != NONE16) {
                        a = (int)bits[(int)e + 1];
                        found = true;
                    }
                    ep[v] = (unsigned short)i;
                    const unsigned short l = lnk[v];
                    v = (l == NONE16) ? -1 : (int)l;
                }
                pred[i] = (signed char)a;
            }
        }
    }
    __syncthreads();

    // ---------------- phase 3: elementwise embedding select ------------------
    const float e0 = emb0[c];
    const float e1 = emb1[c];
    for (int t = tid; t < T_LEN; t += 64) {
        const int a = (int)pred[t];
        const float o = (a < 0) ? 0.0f : (a ? e1 : e0);
        out[((size_t)b * T_LEN + t) * C_DIM + c] = o;
    }
}

extern "C" void kernel_launch(void* const* d_in, const int* in_sizes, int n_in,
                              void* d_out, int out_size, void* d_ws, size_t ws_size,
                              hipStream_t stream) {
    (void)in_sizes; (void)n_in; (void)out_size; (void)d_ws; (void)ws_size;
    const float* x    = (const float*)d_in[0];   // [2,2048,128] f32
    const float* emb0 = (const float*)d_in[1];   // [128] f32 (broadcast [1,1,C])
    const float* emb1 = (const float*)d_in[2];   // [128] f32
    float* out        = (float*)d_out;           // [2,2048,128] f32

    dim3 grid(B_DIM * C_DIM);  // 256 workgroups, one per (b,c) sequence
    dim3 block(64);            // 2 wave32 per workgroup
    hipLaunchKernelGGL(rosa_sa_kernel, grid, block, 0, stream, x, emb0, emb1, out);
}